// MyGaussianPDF_test_1391569403989
// MI455X (gfx1250) — compile-verified
//
#include <hip/hip_runtime.h>
#include <hip/hip_bf16.h>

#ifndef __has_builtin
#define __has_builtin(x) 0
#endif

#if __has_builtin(__builtin_amdgcn_global_load_async_to_lds_b128)
#define USE_ASYNC_LDS 1
#else
#define USE_ASYNC_LDS 0
#endif

#define TPB 256
#define NBLOCKS 2048

#if USE_ASYNC_LDS
// Builtin signature (from hipcc diagnostic): first param is a pointer to
// int __attribute__((vector_size(16))) in the global address space.
typedef int v4i __attribute__((vector_size(16)));
typedef __attribute__((address_space(1))) v4i* gptr_v4i;
typedef __attribute__((address_space(3))) v4i* lptr_v4i;
#endif

// exp(-0.5*q) = exp2(q * (-0.5 * log2(e)))  -> single v_exp_f32
__device__ __forceinline__ float gexp_half_neg(float q) {
    return __builtin_amdgcn_exp2f(q * (-0.72134752044448170368f));
}

__device__ __forceinline__ float quad_form(float px, float py,
                                           float mu0, float mu1,
                                           float c00, float cxy, float c11) {
    float dx = px - mu0;
    float dy = py - mu1;
    // dx*c00*dx + (c01+c10)*dx*dy + dy*c11*dy  (FMA-friendly)
    return fmaf(c00 * dx, dx, fmaf(cxy * dx, dy, c11 * dy * dy));
}

// Main streaming kernel: each thread handles `iters` units of 4 points,
// strided by total thread count. Async global->LDS double buffering on CDNA5.
__global__ void __launch_bounds__(TPB)
gauss_pdf_main(const float4* __restrict__ xv,     // N/2 float4 (2 pts each)
               const float* __restrict__ mu,      // [2]
               const float* __restrict__ cov,     // [2,2]
               float4* __restrict__ outv,         // N/4 float4
               int iters, long long stride_units)
{
    const int tid = threadIdx.x;
    long long u = (long long)blockIdx.x * TPB + tid;   // unit index (4 points)

    const float mu0 = mu[0], mu1 = mu[1];
    const float c00 = cov[0], cxy = cov[1] + cov[2], c11 = cov[3];

#if USE_ASYNC_LDS
    // Per-thread private staging slots: 2 buffers x 32B/thread = 16KB/block.
    __shared__ float4 stage[2][TPB][2];

    // Prologue: stage tile 0 into buffer 0 (two b128 async ops).
    {
        gptr_v4i g = (gptr_v4i)(xv + 2 * u);
        lptr_v4i l = (lptr_v4i)&stage[0][tid][0];
        __builtin_amdgcn_global_load_async_to_lds_b128(g,     l,     0, 0);
        __builtin_amdgcn_global_load_async_to_lds_b128(g + 1, l + 1, 0, 0);
    }

    for (int it = 0; it < iters; ++it) {
        const int buf = it & 1;
        const long long unext = u + stride_units;

        if (it + 1 < iters) {
            // Issue next tile into the other buffer before waiting.
            gptr_v4i g = (gptr_v4i)(xv + 2 * unext);
            lptr_v4i l = (lptr_v4i)&stage[buf ^ 1][tid][0];
            __builtin_amdgcn_global_load_async_to_lds_b128(g,     l,     0, 0);
            __builtin_amdgcn_global_load_async_to_lds_b128(g + 1, l + 1, 0, 0);
            // 4 async ops outstanding; async loads complete in order, so the
            // first 2 (current buffer) are done once the count drops to <=2.
            asm volatile("s_wait_asynccnt 0x2" ::: "memory");
        } else {
            asm volatile("s_wait_asynccnt 0x0" ::: "memory");
        }

        const float4 a = stage[buf][tid][0];   // points 4u, 4u+1
        const float4 b = stage[buf][tid][1];   // points 4u+2, 4u+3

        float4 r;
        r.x = gexp_half_neg(quad_form(a.x, a.y, mu0, mu1, c00, cxy, c11));
        r.y = gexp_half_neg(quad_form(a.z, a.w, mu0, mu1, c00, cxy, c11));
        r.z = gexp_half_neg(quad_form(b.x, b.y, mu0, mu1, c00, cxy, c11));
        r.w = gexp_half_neg(quad_form(b.z, b.w, mu0, mu1, c00, cxy, c11));
        outv[u] = r;

        u = unext;
    }
#else
    // Fallback: direct 128-bit streaming loads + CDNA5 prefetch of next tile.
    for (int it = 0; it < iters; ++it) {
        const long long unext = u + stride_units;
        if (it + 1 < iters) {
            __builtin_prefetch(xv + 2 * unext, 0, 0);      // global_prefetch_b8
        }
        const float4 a = xv[2 * u];
        const float4 b = xv[2 * u + 1];

        float4 r;
        r.x = gexp_half_neg(quad_form(a.x, a.y, mu0, mu1, c00, cxy, c11));
        r.y = gexp_half_neg(quad_form(a.z, a.w, mu0, mu1, c00, cxy, c11));
        r.z = gexp_half_neg(quad_form(b.x, b.y, mu0, mu1, c00, cxy, c11));
        r.w = gexp_half_neg(quad_form(b.z, b.w, mu0, mu1, c00, cxy, c11));
        outv[u] = r;

        u = unext;
    }
#endif
}

// Scalar tail for points not covered by the uniform-trip main kernel.
__global__ void gauss_pdf_tail(const float* __restrict__ x,
                               const float* __restrict__ mu,
                               const float* __restrict__ cov,
                               float* __restrict__ out,
                               long long start, long long n_points)
{
    long long i = start + (long long)blockIdx.x * blockDim.x + threadIdx.x;
    if (i < n_points) {
        const float mu0 = mu[0], mu1 = mu[1];
        const float c00 = cov[0], cxy = cov[1] + cov[2], c11 = cov[3];
        out[i] = gexp_half_neg(quad_form(x[2 * i], x[2 * i + 1],
                                         mu0, mu1, c00, cxy, c11));
    }
}

extern "C" void kernel_launch(void* const* d_in, const int* in_sizes, int n_in,
                              void* d_out, int out_size, void* d_ws, size_t ws_size,
                              hipStream_t stream) {
    (void)n_in; (void)out_size; (void)d_ws; (void)ws_size;

    const float* x   = (const float*)d_in[0];   // [N, 2] f32
    const float* mu  = (const float*)d_in[1];   // [2]    f32
    const float* cov = (const float*)d_in[2];   // [2,2]  f32
    float* out = (float*)d_out;                 // [N]    f32

    const long long n_points = (long long)in_sizes[0] / 2;
    const long long U = n_points / 4;                          // 4-point units
    const long long total_threads = (long long)NBLOCKS * TPB;
    const long long iters = U / total_threads;                 // uniform trips
    const long long umain = iters * total_threads;

    if (iters > 0) {
        gauss_pdf_main<<<NBLOCKS, TPB, 0, stream>>>(
            (const float4*)x, mu, cov, (float4*)out,
            (int)iters, total_threads);
    }

    const long long tail_start = umain * 4;
    const long long tail_n = n_points - tail_start;
    if (tail_n > 0) {
        const int tb = (int)((tail_n + TPB - 1) / TPB);
        gauss_pdf_tail<<<tb, TPB, 0, stream>>>(x, mu, cov, out,
                                               tail_start, n_points);
    }
}